// SpatialAttentionLoss_83683142795672
// MI455X (gfx1250) — compile-verified
//
#include <hip/hip_runtime.h>

// SpatialAttentionLoss for MI455X (gfx1250, wave32).
// Stage 1: stream 4x64MiB f32 tensors with non-temporal b128 loads, accumulate
//          6 per-cell partial sums, cross-lane reduce via V_WMMA_F32_16X16X4_F32
//          (B = ones), cross-wave reduce via LDS, write per-block partials.
// Stage 2: tiny kernel folds partials, applies presence mask / absence /
//          confidence terms, writes the weighted scalar total.

typedef float v4f __attribute__((ext_vector_type(4)));
typedef float v2f __attribute__((ext_vector_type(2)));
typedef float v8f __attribute__((ext_vector_type(8)));

#define N_CELLS 64            // B*S = 8*8
#define BPC 64                // blocks per cell
#define VEC_PER_BLOCK 1024    // float4 per block  (65536 float4/cell / 64)
#define ITERS 4               // 1024 / 256 threads
#define INV_PIX (1.0f / 262144.0f)

// Full 32-lane sum on the matrix pipe: A(16x4) = {v, 0} per lane, B = ones(4x16).
// rowsum(m) = v[m] + v[m+16]; D row m replicated over N. Lanes 0-15 then hold
// sum(rows 0..7), lanes 16-31 hold sum(rows 8..15); two broadcasts finish.
__device__ __forceinline__ float wave_sum_wmma(float v) {
    v2f a; a[0] = v;    a[1] = 0.0f;
    v2f b; b[0] = 1.0f; b[1] = 1.0f;        // B = all ones -> layout-agnostic
    v8f c = {};
    v8f d = __builtin_amdgcn_wmma_f32_16x16x4_f32(
        /*neg_a=*/false, a, /*neg_b=*/false, b,
        /*c_mod=*/(short)0, c, /*reuse_a=*/false, /*reuse_b=*/false);
    float s = d[0] + d[1] + d[2] + d[3] + d[4] + d[5] + d[6] + d[7];
    return __shfl(s, 0, 32) + __shfl(s, 16, 32);
}

__global__ __launch_bounds__(256) void sal_partials(
    const float* __restrict__ seg_logits,
    const float* __restrict__ seg_probs,
    const float* __restrict__ seg_targets,
    const float* __restrict__ att_maps,
    float* __restrict__ partials)           // [6][N_CELLS][BPC]
{
    const int cell = blockIdx.x >> 6;
    const int blk  = blockIdx.x & 63;
    const long base = (long)cell * 65536 + (long)blk * VEC_PER_BLOCK; // float4 units

    const v4f* X = (const v4f*)seg_logits;
    const v4f* P = (const v4f*)seg_probs;
    const v4f* T = (const v4f*)seg_targets;
    const v4f* A = (const v4f*)att_maps;

    float acc[6] = {0.f, 0.f, 0.f, 0.f, 0.f, 0.f};

    #pragma unroll
    for (int it = 0; it < ITERS; ++it) {
        const long idx = base + it * 256 + threadIdx.x;
        v4f x4 = __builtin_nontemporal_load(X + idx);
        v4f p4 = __builtin_nontemporal_load(P + idx);
        v4f t4 = __builtin_nontemporal_load(T + idx);
        v4f a4 = __builtin_nontemporal_load(A + idx);
        #pragma unroll
        for (int j = 0; j < 4; ++j) {
            const float x = x4[j], p = p4[j], t = t4[j], a = a4[j];
            // stable BCE-with-logits
            const float soft = __logf(1.0f + __expf(-fabsf(x)));
            const float bce  = fmaxf(x, 0.0f) - x * t + soft;
            const float pt   = __expf(-bce);
            const float om   = 1.0f - pt;
            acc[0] += bce;                        // seg BCE
            acc[1] += 0.25f * om * om * bce;      // focal (alpha=0.25, gamma=2)
            acc[2] += p * t;                      // dice intersection
            acc[3] += p;                          // dice denom (probs)
            acc[4] += t;                          // dice denom (targets)
            // attention BCE vs (t > 0.5), log terms clamped at -100
            const float lp = fmaxf(__logf(a),        -100.0f);
            const float lq = fmaxf(__logf(1.0f - a), -100.0f);
            acc[5] -= (t > 0.5f) ? lp : lq;
        }
    }

    __shared__ float wsum[8][6];
    const int wid  = threadIdx.x >> 5;
    const int lane = threadIdx.x & 31;
    #pragma unroll
    for (int q = 0; q < 6; ++q) {
        const float s = wave_sum_wmma(acc[q]);    // EXEC all-ones here
        if (lane == 0) wsum[wid][q] = s;
    }
    __syncthreads();
    if (threadIdx.x < 6) {
        float s = 0.0f;
        #pragma unroll
        for (int w = 0; w < 8; ++w) s += wsum[w][threadIdx.x];
        partials[threadIdx.x * (N_CELLS * BPC) + cell * BPC + blk] = s;
    }
}

__global__ __launch_bounds__(64) void sal_finalize(
    const float* __restrict__ partials,       // [6][N_CELLS][BPC]
    const float* __restrict__ presence_probs, // [64]
    const int*   __restrict__ presence_targets,
    float* __restrict__ out)
{
    const int c = threadIdx.x;                // one thread per (b,s) cell
    float s[6] = {0.f, 0.f, 0.f, 0.f, 0.f, 0.f};
    for (int blk = 0; blk < BPC; ++blk) {
        #pragma unroll
        for (int q = 0; q < 6; ++q)
            s[q] += partials[q * (N_CELLS * BPC) + c * BPC + blk];
    }

    const float bce_mean   = s[0] * INV_PIX;
    const float focal_mean = s[1] * INV_PIX;
    const float dice       = 1.0f - (2.0f * s[2] + 1e-6f) / (s[3] + s[4] + 1e-6f);
    const float att_mean   = s[5] * INV_PIX;

    const float pres = (float)presence_targets[c];
    const float qp   = presence_probs[c];
    const float lp   = fmaxf(__logf(qp),        -100.0f);
    const float lq   = fmaxf(__logf(1.0f - qp), -100.0f);
    const float absence = -(pres * lp + (1.0f - pres) * lq);
    const bool wrong = ((pres == 0.0f) && (qp > 0.5f)) ||
                       ((pres == 1.0f) && (qp < 0.5f));
    const float dq   = qp - 0.5f;
    const float conf = wrong ? dq * dq : 0.0f;

    float r[7] = { pres, bce_mean * pres, dice * pres, focal_mean * pres,
                   att_mean * pres, absence, conf };
    #pragma unroll
    for (int q = 0; q < 7; ++q)
        for (int off = 16; off > 0; off >>= 1)
            r[q] += __shfl_down(r[q], off, 32);

    __shared__ float red[2][7];
    if ((threadIdx.x & 31) == 0) {
        #pragma unroll
        for (int q = 0; q < 7; ++q) red[threadIdx.x >> 5][q] = r[q];
    }
    __syncthreads();
    if (threadIdx.x == 0) {
        const float cnt   = red[0][0] + red[1][0];
        const float seg   = red[0][1] + red[1][1];
        const float dic   = red[0][2] + red[1][2];
        const float foc   = red[0][3] + red[1][3];
        const float att   = red[0][4] + red[1][4];
        const float abs_s = red[0][5] + red[1][5];
        const float con   = red[0][6] + red[1][6];
        const float inv_cnt = (cnt > 0.0f) ? 1.0f / cnt : 0.0f;
        const float inv_n   = 1.0f / 64.0f;   // 1 / (B*S)
        // weights: seg 1.0, dice 1.0, focal 0.5, absence 1.0, attention 0.5, conf 0.1
        out[0] = (seg + dic) * inv_cnt + 0.5f * foc * inv_cnt
               + abs_s * inv_n + 0.5f * att * inv_n + 0.1f * con * inv_n;
    }
}

extern "C" void kernel_launch(void* const* d_in, const int* in_sizes, int n_in,
                              void* d_out, int out_size, void* d_ws, size_t ws_size,
                              hipStream_t stream) {
    const float* seg_logits       = (const float*)d_in[0];
    const float* seg_probs        = (const float*)d_in[1];
    const float* seg_targets      = (const float*)d_in[2];
    const float* presence_probs   = (const float*)d_in[3];
    const float* attention_maps   = (const float*)d_in[4];
    const int*   presence_targets = (const int*)d_in[5];
    float* partials = (float*)d_ws;           // 6*64*64 floats = 96 KB

    sal_partials<<<N_CELLS * BPC, 256, 0, stream>>>(
        seg_logits, seg_probs, seg_targets, attention_maps, partials);
    sal_finalize<<<1, 64, 0, stream>>>(
        partials, presence_probs, presence_targets, (float*)d_out);
}